// BiInteraction_66090956751485
// MI455X (gfx1250) — compile-verified
//
#include <hip/hip_runtime.h>

#define A_N    2048
#define L_N    512
#define DA_N   128
#define DP_N   128
#define B_N    64
#define H1_N   512
#define H2_N   256

typedef __attribute__((ext_vector_type(2))) float v2f;
typedef __attribute__((ext_vector_type(8))) float v8f;

// D = A(16x4,f32) * B(4x16,f32) + C(16x16,f32)  -> v_wmma_f32_16x16x4_f32
__device__ __forceinline__ v8f wmma4(v2f a, v2f b, v8f c) {
  return __builtin_amdgcn_wmma_f32_16x16x4_f32(
      /*neg_a=*/false, a, /*neg_b=*/false, b,
      /*c_mod=*/(short)0, c, /*reuse_a=*/false, /*reuse_b=*/false);
}

// float atomic-max via monotone int mapping (works in LDS: ds_max_i32/ds_min_u32)
__device__ __forceinline__ void atomicMaxF(float* addr, float v) {
  if (v >= 0.0f) atomicMax((int*)addr, __float_as_int(v));
  else           atomicMin((unsigned int*)addr, __float_as_uint(v));
}

// ---------------- Kernel 1: X = atom_embed @ W_att  (2048x128 @ 128x128) ----
__global__ void k_gemm_att(const float* __restrict__ atomE,
                           const float* __restrict__ Watt,
                           float* __restrict__ X) {
  const int wave = blockIdx.x * (blockDim.x >> 5) + (threadIdx.x >> 5);
  const int lane = threadIdx.x & 31;
  const int hi = lane >> 4, ln = lane & 15;
  const int tilesN = DP_N >> 4;                 // 8
  const int tm = wave / tilesN, tn = wave % tilesN;
  const int arow = tm * 16 + ln;                // A-matrix M index (both halves)
  const int col  = tn * 16 + ln;                // B/C/D N index
  v8f c = {};
  for (int k = 0; k < DA_N; k += 4) {
    const int k0 = k + (hi << 1);
    v2f a, b;
    a.x = atomE[arow * DA_N + k0];
    a.y = atomE[arow * DA_N + k0 + 1];
    b.x = Watt[k0 * DP_N + col];
    b.y = Watt[(k0 + 1) * DP_N + col];
    c = wmma4(a, b, c);
  }
#pragma unroll
  for (int i = 0; i < 8; ++i) {
    const int m = tm * 16 + i + (hi << 3);
    X[m * DP_N + col] = c[i];
  }
}

// ---------------- Kernel 2: per-molecule scores, maxes, pools ---------------
// grid = B_N blocks, 256 threads (8 waves). Atoms sorted by segment, so the
// score block S_b = X_b @ P_b^T is a dense per-molecule WMMA GEMM.
__global__ void k_segment(const float* __restrict__ X,
                          const float* __restrict__ P,
                          const float* __restrict__ atomE,
                          const int* __restrict__ splits,
                          float* __restrict__ WcA,
                          float* __restrict__ h) {
  const int b   = blockIdx.x;
  const int tid = threadIdx.x;
  __shared__ float colmax[L_N];   // segment-max over atoms, per residue
  __shared__ float rowmax[16];    // max over residues, per atom (current tile)
  __shared__ float red[256];
  __shared__ float sSc;

  const float NEG_INF = -__builtin_inff();
  for (int i = tid; i < L_N; i += 256) colmax[i] = NEG_INF;
  if (tid == 0) sSc = 0.0f;

  // atom range of molecule b via binary search in sorted splits
  int lo = 0, hib = A_N;
  while (lo < hib) { int mid = (lo + hib) >> 1; if (splits[mid] < b) lo = mid + 1; else hib = mid; }
  const int start = lo;
  lo = start; hib = A_N;
  while (lo < hib) { int mid = (lo + hib) >> 1; if (splits[mid] < b + 1) lo = mid + 1; else hib = mid; }
  const int end = lo;
  const int nb  = end - start;
  __syncthreads();

  if (nb <= 0) {  // empty segment: emit zeros, uniform early-out
    if (tid < DA_N + DP_N) h[b * (DA_N + DP_N) + tid] = 0.0f;
    return;
  }

  const int wv = tid >> 5;
  const int lane = tid & 31;
  const int hi = lane >> 4, ln = lane & 15;
  const float* Pb = P + (size_t)b * L_N * DP_N;
  const int nrt = (nb + 15) >> 4;

  for (int rt = 0; rt < nrt; ++rt) {
    if (tid < 16) rowmax[tid] = NEG_INF;
    __syncthreads();
    int arow = start + rt * 16 + ln;            // tile row (clamped; masked below)
    if (arow > A_N - 1) arow = A_N - 1;
    for (int ct = wv; ct < (L_N >> 4); ct += 8) {   // 4 col-tiles per wave
      const int colbase = ct << 4;
      v8f c = {};
      for (int k = 0; k < DP_N; k += 4) {
        const int k0 = k + (hi << 1);
        v2f a, bb;
        a.x = X[arow * DP_N + k0];
        a.y = X[arow * DP_N + k0 + 1];
        const float* pl = Pb + (size_t)(colbase + ln) * DP_N + k0;
        bb.x = pl[0];
        bb.y = pl[1];
        c = wmma4(a, bb, c);
      }
#pragma unroll
      for (int i = 0; i < 8; ++i) {
        const int m = i + (hi << 3);
        if (start + rt * 16 + m < end) {        // mask padded rows
          atomicMaxF(&rowmax[m], c[i]);
          atomicMaxF(&colmax[colbase + ln], c[i]);
        }
      }
    }
    __syncthreads();
    if (tid < 16) {
      const int atom = start + rt * 16 + tid;
      if (atom < end) {
        const float wc = __expf(rowmax[tid]);   // Wc = exp(max_l S)
        WcA[atom] = wc;
        atomicAdd(&sSc, wc);
      }
    }
    __syncthreads();
  }

  // softmax over residues of Wp = colmax[0..511]
  float mx = NEG_INF;
  for (int i = tid; i < L_N; i += 256) mx = fmaxf(mx, colmax[i]);
  red[tid] = mx; __syncthreads();
  for (int s = 128; s > 0; s >>= 1) { if (tid < s) red[tid] = fmaxf(red[tid], red[tid + s]); __syncthreads(); }
  const float gmax = red[0];
  __syncthreads();
  float ssum = 0.0f;
  for (int i = tid; i < L_N; i += 256) { const float e = __expf(colmax[i] - gmax); colmax[i] = e; ssum += e; }
  red[tid] = ssum; __syncthreads();
  for (int s = 128; s > 0; s >>= 1) { if (tid < s) red[tid] += red[tid + s]; __syncthreads(); }
  const float denom = red[0];
  __syncthreads();

  if (tid < DP_N) {                             // prot_pool -> h[:,128:256]
    float acc = 0.0f;
    for (int l = 0; l < L_N; ++l) acc += colmax[l] * Pb[(size_t)l * DP_N + tid];
    h[b * (DA_N + DP_N) + DA_N + tid] = acc / denom;
  } else if (tid - 128 < DA_N) {                // atom_pool -> h[:,0:128]
    const int d = tid - 128;
    float acc = 0.0f;
    for (int a = start; a < end; ++a) acc += WcA[a] * atomE[a * DA_N + d];
    h[b * (DA_N + DP_N) + d] = acc / sSc;
  }
}

// ---------------- Kernel 3: generic WMMA GEMM + bias (+ReLU) ---------------
__global__ void k_gemm_mlp(const float* __restrict__ Ain, const float* __restrict__ W,
                           const float* __restrict__ bias, float* __restrict__ Out,
                           const int M, const int K, const int N, const int doRelu) {
  const int wave = blockIdx.x * (blockDim.x >> 5) + (threadIdx.x >> 5);
  const int lane = threadIdx.x & 31;
  const int hi = lane >> 4, ln = lane & 15;
  const int tilesN = N >> 4;
  const int tm = wave / tilesN, tn = wave % tilesN;
  if (tm * 16 >= M) return;                     // wave-uniform
  const int row = tm * 16 + ln;
  const int col = tn * 16 + ln;
  v8f c = {};
  for (int k = 0; k < K; k += 4) {
    const int k0 = k + (hi << 1);
    v2f a, b;
    a.x = Ain[row * K + k0];
    a.y = Ain[row * K + k0 + 1];
    b.x = W[k0 * N + col];
    b.y = W[(k0 + 1) * N + col];
    c = wmma4(a, b, c);
  }
  const float bv = bias[col];
#pragma unroll
  for (int i = 0; i < 8; ++i) {
    const int m = tm * 16 + i + (hi << 3);
    float v = c[i] + bv;
    if (doRelu) v = fmaxf(v, 0.0f);
    Out[m * N + col] = v;
  }
}

// ---------------- Kernel 4: out = h2 @ Wo + bo ------------------------------
__global__ void k_out(const float* __restrict__ h2, const float* __restrict__ Wo,
                      const float* __restrict__ bo, float* __restrict__ out) {
  const int b = threadIdx.x;
  if (b < B_N) {
    float acc = bo[0];
    for (int k = 0; k < H2_N; ++k) acc += h2[b * H2_N + k] * Wo[k];
    out[b] = acc;
  }
}

extern "C" void kernel_launch(void* const* d_in, const int* in_sizes, int n_in,
                              void* d_out, int out_size, void* d_ws, size_t ws_size,
                              hipStream_t stream) {
  (void)in_sizes; (void)n_in; (void)out_size; (void)ws_size;
  const float* atomE  = (const float*)d_in[0];
  const float* P      = (const float*)d_in[1];
  const int*   splits = (const int*)d_in[2];
  const float* Watt   = (const float*)d_in[3];
  const float* W1     = (const float*)d_in[4];
  const float* b1     = (const float*)d_in[5];
  const float* W2     = (const float*)d_in[6];
  const float* b2     = (const float*)d_in[7];
  const float* Wo     = (const float*)d_in[8];
  const float* bo     = (const float*)d_in[9];
  float* out = (float*)d_out;

  float* X   = (float*)d_ws;                    // 2048*128
  float* WcA = X + A_N * DP_N;                  // 2048
  float* h   = WcA + A_N;                       // 64*256
  float* h1  = h + B_N * (DA_N + DP_N);         // 64*512
  float* h2  = h1 + B_N * H1_N;                 // 64*256

  // 1) X = atomE @ W_att : 128x8 tiles -> 1024 waves, 8 waves/block
  k_gemm_att<<<dim3((A_N / 16) * (DP_N / 16) / 8), dim3(256), 0, stream>>>(atomE, Watt, X);
  // 2) per-molecule scores + pools
  k_segment<<<dim3(B_N), dim3(256), 0, stream>>>(X, P, atomE, splits, WcA, h);
  // 3) MLP: h[64,256] -> h1[64,512] -> h2[64,256]
  k_gemm_mlp<<<dim3((B_N / 16) * (H1_N / 16) / 8), dim3(256), 0, stream>>>(
      h, W1, b1, h1, B_N, DA_N + DP_N, H1_N, 1);
  k_gemm_mlp<<<dim3((B_N / 16) * (H2_N / 16) / 8), dim3(256), 0, stream>>>(
      h1, W2, b2, h2, B_N, H1_N, H2_N, 1);
  // 4) final projection
  k_out<<<dim3(1), dim3(64), 0, stream>>>(h2, Wo, bo, out);
}